// PoolLayer_41953240547506
// MI455X (gfx1250) — compile-verified
//
#include <hip/hip_runtime.h>
#include <math.h>

typedef float v2f __attribute__((ext_vector_type(2)));
typedef float v8f __attribute__((ext_vector_type(8)));

// F value for a member row: (1 - 1 + 1e-6) / (1 - 1e-6)
#define FM ((1.0f - 1.0f + 1e-6f) / (1.0f - 1e-6f))

#define N_ATOMS 4096
#define KB 16
#define GG 64

// ---------------- bond MLP: [N*K,16] -> scalar (16->6 relu ->6 relu ->1) ----
__global__ void k_bond_mlp(const float* __restrict__ bf,
                           const float* __restrict__ w0, const float* __restrict__ b0,
                           const float* __restrict__ w1, const float* __restrict__ b1,
                           const float* __restrict__ w2, const float* __restrict__ b2,
                           float* __restrict__ bond) {
  int idx = blockIdx.x * blockDim.x + threadIdx.x;     // 0..65535
  const float* x = bf + (size_t)idx * 16;
  float xr[16];
#pragma unroll
  for (int i = 0; i < 16; ++i) xr[i] = x[i];
  float h0[6];
#pragma unroll
  for (int j = 0; j < 6; ++j) {
    float a = b0[j];
#pragma unroll
    for (int i = 0; i < 16; ++i) a += xr[i] * w0[i * 6 + j];
    h0[j] = fmaxf(a, 0.f);
  }
  float h1[6];
#pragma unroll
  for (int j = 0; j < 6; ++j) {
    float a = b1[j];
#pragma unroll
    for (int i = 0; i < 6; ++i) a += h0[i] * w1[i * 6 + j];
    h1[j] = fmaxf(a, 0.f);
  }
  float o = b2[0];
#pragma unroll
  for (int i = 0; i < 6; ++i) o += h1[i] * w2[i];
  bond[idx] = o;
}

// ---------------- group id from one-hot M ----------------------------------
__global__ void k_gid(const float* __restrict__ M, int* __restrict__ gid) {
  int n = blockIdx.x * blockDim.x + threadIdx.x;
  const float* row = M + (size_t)n * GG;
  int g = 0;
  for (int j = 0; j < GG; ++j)
    if (row[j] > 0.5f) g = j;
  gid[n] = g;
}

// ---------------- per-group bond pooling (Ms_bond1 + Ms_bond2 fused) -------
// Produces bp2[g*64+g2]: diag = softmax-weighted sum of bp, off-diag = mean.
__global__ void k_bond_group(const float* __restrict__ bond, const int* __restrict__ gid,
                             float* __restrict__ bp2) {
  __shared__ int s_cnt;
  __shared__ int s_mem[N_ATOMS];
  __shared__ float s_bp[N_ATOMS];
  __shared__ float s_mx[KB], s_sum[KB];
  __shared__ float s_mean, s_diag;
  int g = blockIdx.x, tid = threadIdx.x;
  if (tid == 0) s_cnt = 0;
  __syncthreads();
  for (int n = tid; n < N_ATOMS; n += blockDim.x)
    if (gid[n] == g) { int p = atomicAdd(&s_cnt, 1); s_mem[p] = n; }
  __syncthreads();
  int cnt = s_cnt;
  if (tid < KB) {                       // per-k softmax stats over members
    int k = tid;
    float mx = -INFINITY;
    for (int i = 0; i < cnt; ++i) mx = fmaxf(mx, bond[s_mem[i] * KB + k] + FM);
    float s = 0.f;
    for (int i = 0; i < cnt; ++i) s += expf(bond[s_mem[i] * KB + k] + FM - mx);
    s_mx[k] = mx; s_sum[k] = s;
  }
  __syncthreads();
  for (int i = tid; i < cnt; i += blockDim.x) {   // bp[n] = sum_k bond * Ms1
    int n = s_mem[i];
    float acc = 0.f;
    for (int k = 0; k < KB; ++k) {
      float b = bond[n * KB + k];
      acc += b * (expf(b + FM - s_mx[k]) / s_sum[k]);
    }
    s_bp[i] = acc;
  }
  __syncthreads();
  if (tid == 0) {
    float mean = 0.f, diag = 0.f;
    if (cnt > 0) {
      float mx = -INFINITY;
      for (int i = 0; i < cnt; ++i) mx = fmaxf(mx, s_bp[i] + FM);
      float s = 0.f, sb = 0.f, sw = 0.f;
      for (int i = 0; i < cnt; ++i) {
        float e = expf(s_bp[i] + FM - mx);
        s += e; sb += s_bp[i]; sw += s_bp[i] * e;
      }
      mean = sb / (float)cnt;
      diag = sw / s;
    }
    s_mean = mean; s_diag = diag;
  }
  __syncthreads();
  for (int g2 = tid; g2 < GG; g2 += blockDim.x)
    bp2[g * GG + g2] = (g2 == g) ? s_diag : s_mean;
}

// ---------------- p2 MLP front half: scalar -> 64 (relu) -------------------
__global__ void k_p2_h1(const float* __restrict__ bp2,
                        const float* __restrict__ w0, const float* __restrict__ b0,
                        const float* __restrict__ w1, const float* __restrict__ b1,
                        float* __restrict__ h1buf) {
  int r = blockIdx.x * blockDim.x + threadIdx.x;   // 0..4095
  float x = bp2[r];
  float h0 = fmaxf(x * w0[0] + b0[0], 0.f);
  float* out = h1buf + (size_t)r * 64;
#pragma unroll 4
  for (int j = 0; j < 64; ++j) out[j] = fmaxf(h0 * w1[j] + b1[j], 0.f);
}

// ---------------- WMMA GEMM: [4096,64] x [64,128] + bias -> bond_pool ------
__global__ void __launch_bounds__(32) k_gemm_h1w2(
    const float* __restrict__ A,   // [4096,64]
    const float* __restrict__ W,   // [64,128]
    const float* __restrict__ B,   // [128]
    float* __restrict__ C) {       // [4096,128]
  int bid = blockIdx.x;
  int ct = bid & 7, rt = bid >> 3;
  int lane = threadIdx.x;
  int l16 = lane & 15, half = lane >> 4;
  int rbase = rt * 16, cbase = ct * 16;
  v8f acc = {0.f, 0.f, 0.f, 0.f, 0.f, 0.f, 0.f, 0.f};
  const float* arow = A + (size_t)(rbase + l16) * 64 + 2 * half;
#pragma unroll
  for (int kk = 0; kk < 64; kk += 4) {
    v2f a; a.x = arow[kk]; a.y = arow[kk + 1];               // A: M=l16, K=kk+2h+{0,1}
    int kr = kk + 2 * half;
    v2f b;                                                    // B: N=l16, K=kr+{0,1}
    b.x = W[kr * 128 + cbase + l16];
    b.y = W[(kr + 1) * 128 + cbase + l16];
    acc = __builtin_amdgcn_wmma_f32_16x16x4_f32(false, a, false, b,
                                                (short)0, acc, false, false);
  }
  float bias = B[cbase + l16];
#pragma unroll
  for (int j = 0; j < 8; ++j) {
    int m = j + 8 * half;                                     // D: M=j+8h, N=l16
    C[(size_t)(rbase + m) * 128 + cbase + l16] = acc[j] + bias;
  }
}

// ---------------- q/k/v MLP: 64 -> 16 relu -> 24 relu -> 32 ----------------
__global__ void k_mlp_qkv(const float* __restrict__ X,
                          const float* __restrict__ w0, const float* __restrict__ b0,
                          const float* __restrict__ w1, const float* __restrict__ b1,
                          const float* __restrict__ w2, const float* __restrict__ b2,
                          float* __restrict__ Y) {
  int n = blockIdx.x * blockDim.x + threadIdx.x;
  const float* x = X + (size_t)n * 64;
  float h0[16];
  for (int j = 0; j < 16; ++j) {
    float a = b0[j];
    for (int i = 0; i < 64; ++i) a += x[i] * w0[i * 16 + j];
    h0[j] = fmaxf(a, 0.f);
  }
  float h1[24];
  for (int j = 0; j < 24; ++j) {
    float a = b1[j];
#pragma unroll
    for (int i = 0; i < 16; ++i) a += h0[i] * w1[i * 24 + j];
    h1[j] = fmaxf(a, 0.f);
  }
  float* y = Y + (size_t)n * 32;
  for (int j = 0; j < 32; ++j) {
    float a = b2[j];
#pragma unroll
    for (int i = 0; i < 24; ++i) a += h1[i] * w2[i * 32 + j];
    y[j] = a;
  }
}

// ---------------- flash attention, 8 heads, hd=4, WMMA f32 16x16x4 ---------
// One wave per (head, 16-query tile). 4 key tiles (64 keys) per online-softmax
// round: 4 WMMAs produce 4 score tiles; per query row the 4 per-tile values
// are combined in-register (3 v_max / 3 v_add) so the 4-level shfl_xor
// reduction tree runs once per 64 keys instead of once per 16.
// D layout: lane holds S[m=j+8*half, n=l16]. P*V kept as per-lane partials
// (lane owns key column n=l16 of each tile), reduced across lanes at the end.
__global__ void __launch_bounds__(32) k_attn(
    const float* __restrict__ Q, const float* __restrict__ K,
    const float* __restrict__ V, float* __restrict__ O) {
  int bid = blockIdx.x;          // 2048 = 8 heads * 256 query tiles
  int h = bid >> 8;
  int qt = bid & 255;
  int lane = threadIdx.x;
  int l16 = lane & 15, half = lane >> 4;
  int qbase = qt * 16;

  const float* qp = Q + (size_t)(qbase + l16) * 32 + h * 4 + 2 * half;
  v2f a; a.x = qp[0]; a.y = qp[1];          // A: M=l16(query), K=2h+{0,1}

  float m_run[8], l_run[8], o_part[8][4];
#pragma unroll
  for (int j = 0; j < 8; ++j) {
    m_run[j] = -INFINITY; l_run[j] = 0.f;
#pragma unroll
    for (int d = 0; d < 4; ++d) o_part[j][d] = 0.f;
  }

  for (int kt = 0; kt < 64; ++kt) {        // 64 iterations x 64 keys
    int kbase = kt * 64;
    v2f b[4];
    float4 vr[4];
#pragma unroll
    for (int t = 0; t < 4; ++t) {
      int krow = kbase + t * 16 + l16;
      const float* kp = K + (size_t)krow * 32 + h * 4 + 2 * half;
      b[t].x = kp[0]; b[t].y = kp[1];      // B: N=l16(key), K=2h+{0,1}
      vr[t] = *(const float4*)(V + (size_t)krow * 32 + h * 4);
    }
    v8f c[4];
#pragma unroll
    for (int t = 0; t < 4; ++t) {
      v8f z = {0.f, 0.f, 0.f, 0.f, 0.f, 0.f, 0.f, 0.f};
      c[t] = __builtin_amdgcn_wmma_f32_16x16x4_f32(false, a, false, b[t],
                                                   (short)0, z, false, false);
    }
#pragma unroll
    for (int j = 0; j < 8; ++j) {
      float s0 = c[0][j] * 0.5f;           // 1/sqrt(head_dim)
      float s1 = c[1][j] * 0.5f;
      float s2 = c[2][j] * 0.5f;
      float s3 = c[3][j] * 0.5f;
      float mx = fmaxf(fmaxf(s0, s1), fmaxf(s2, s3));
      mx = fmaxf(mx, __shfl_xor(mx, 1, 32));
      mx = fmaxf(mx, __shfl_xor(mx, 2, 32));
      mx = fmaxf(mx, __shfl_xor(mx, 4, 32));
      mx = fmaxf(mx, __shfl_xor(mx, 8, 32));
      float m_new = fmaxf(m_run[j], mx);
      float corr = expf(m_run[j] - m_new); // 0 on first iteration (exp(-inf))
      float p0 = expf(s0 - m_new);
      float p1 = expf(s1 - m_new);
      float p2 = expf(s2 - m_new);
      float p3 = expf(s3 - m_new);
      float ps = (p0 + p1) + (p2 + p3);
      ps += __shfl_xor(ps, 1, 32);
      ps += __shfl_xor(ps, 2, 32);
      ps += __shfl_xor(ps, 4, 32);
      ps += __shfl_xor(ps, 8, 32);
      l_run[j] = l_run[j] * corr + ps;
      m_run[j] = m_new;
      o_part[j][0] = o_part[j][0] * corr + p0 * vr[0].x + p1 * vr[1].x + p2 * vr[2].x + p3 * vr[3].x;
      o_part[j][1] = o_part[j][1] * corr + p0 * vr[0].y + p1 * vr[1].y + p2 * vr[2].y + p3 * vr[3].y;
      o_part[j][2] = o_part[j][2] * corr + p0 * vr[0].z + p1 * vr[1].z + p2 * vr[2].z + p3 * vr[3].z;
      o_part[j][3] = o_part[j][3] * corr + p0 * vr[0].w + p1 * vr[1].w + p2 * vr[2].w + p3 * vr[3].w;
    }
  }
  // reduce partial outputs over the 16 key lanes, normalize, store
#pragma unroll
  for (int j = 0; j < 8; ++j) {
    float inv = 1.f / l_run[j];
#pragma unroll
    for (int d = 0; d < 4; ++d) {
      float v = o_part[j][d];
      v += __shfl_xor(v, 1, 32);
      v += __shfl_xor(v, 2, 32);
      v += __shfl_xor(v, 4, 32);
      v += __shfl_xor(v, 8, 32);
      o_part[j][d] = v * inv;
    }
  }
  if (l16 == 0) {
#pragma unroll
    for (int j = 0; j < 8; ++j) {
      int m = j + 8 * half;
      float* op = O + (size_t)(qbase + m) * 32 + h * 4;
      op[0] = o_part[j][0]; op[1] = o_part[j][1];
      op[2] = o_part[j][2]; op[3] = o_part[j][3];
    }
  }
}

// ---------------- p3 + p4 head: 32->24r->16r->11, 11->11r->11r->21 ---------
__global__ void k_atom_head(const float* __restrict__ X,
    const float* __restrict__ w30, const float* __restrict__ b30,
    const float* __restrict__ w31, const float* __restrict__ b31,
    const float* __restrict__ w32, const float* __restrict__ b32,
    const float* __restrict__ w40, const float* __restrict__ b40,
    const float* __restrict__ w41, const float* __restrict__ b41,
    const float* __restrict__ w42, const float* __restrict__ b42,
    float* __restrict__ Y) {
  int n = blockIdx.x * blockDim.x + threadIdx.x;
  const float* x = X + (size_t)n * 32;
  float h[24];
  for (int j = 0; j < 24; ++j) {
    float aa = b30[j];
    for (int i = 0; i < 32; ++i) aa += x[i] * w30[i * 24 + j];
    h[j] = fmaxf(aa, 0.f);
  }
  float h2[16];
  for (int j = 0; j < 16; ++j) {
    float aa = b31[j];
#pragma unroll
    for (int i = 0; i < 24; ++i) aa += h[i] * w31[i * 16 + j];
    h2[j] = fmaxf(aa, 0.f);
  }
  float h3[11];
  for (int j = 0; j < 11; ++j) {
    float aa = b32[j];
#pragma unroll
    for (int i = 0; i < 16; ++i) aa += h2[i] * w32[i * 11 + j];
    h3[j] = aa;                                  // no relu (last of p3)
  }
  float h4[11];
  for (int j = 0; j < 11; ++j) {
    float aa = b40[j];
#pragma unroll
    for (int i = 0; i < 11; ++i) aa += h3[i] * w40[i * 11 + j];
    h4[j] = fmaxf(aa, 0.f);
  }
  float h5[11];
  for (int j = 0; j < 11; ++j) {
    float aa = b41[j];
#pragma unroll
    for (int i = 0; i < 11; ++i) aa += h4[i] * w41[i * 11 + j];
    h5[j] = fmaxf(aa, 0.f);
  }
  float* y = Y + (size_t)n * 21;
  for (int j = 0; j < 21; ++j) {
    float aa = b42[j];
#pragma unroll
    for (int i = 0; i < 11; ++i) aa += h5[i] * w42[i * 21 + j];
    y[j] = aa;                                   // no relu (last of p4)
  }
}

// ---------------- per-group atom pooling -----------------------------------
__global__ void k_atom_pool(const float* __restrict__ A21, const int* __restrict__ gid,
                            float* __restrict__ out) {
  __shared__ int s_cnt;
  __shared__ int s_mem[N_ATOMS];
  int g = blockIdx.x, tid = threadIdx.x;
  if (tid == 0) s_cnt = 0;
  __syncthreads();
  for (int n = tid; n < N_ATOMS; n += blockDim.x)
    if (gid[n] == g) { int p = atomicAdd(&s_cnt, 1); s_mem[p] = n; }
  __syncthreads();
  int cnt = s_cnt;
  if (tid < 21) {
    int c = tid;
    float mx = -INFINITY;
    for (int i = 0; i < cnt; ++i) mx = fmaxf(mx, A21[s_mem[i] * 21 + c] + FM);
    float s = 0.f, sw = 0.f;
    for (int i = 0; i < cnt; ++i) {
      float av = A21[s_mem[i] * 21 + c];
      float e = expf(av + FM - mx);
      s += e; sw += av * e;
    }
    out[g * 21 + c] = (cnt > 0) ? (sw / s) : 0.f;
  }
}

extern "C" void kernel_launch(void* const* d_in, const int* in_sizes, int n_in,
                              void* d_out, int out_size, void* d_ws, size_t ws_size,
                              hipStream_t stream) {
  (void)in_sizes; (void)n_in; (void)out_size; (void)ws_size;
  const float* bond_feat = (const float*)d_in[0];
  const float* atom_feat = (const float*)d_in[1];
  const float* Mmat      = (const float*)d_in[2];
  auto F = [&](int i) { return (const float*)d_in[i]; };

  char* ws = (char*)d_ws;
  float* bond  = (float*)(ws + 0);          // 65536 f  (256 KiB)
  int*   gid   = (int*)  (ws + 262144);     // 4096 i32 (16 KiB)
  float* bp2   = (float*)(ws + 278528);     // 4096 f   (16 KiB)
  float* h1buf = (float*)(ws + 294912);     // 262144 f (1 MiB)
  float* qb    = (float*)(ws + 1343488);    // 131072 f (512 KiB)
  float* kb    = (float*)(ws + 1867776);    // 131072 f
  float* vb    = (float*)(ws + 2392064);    // 131072 f
  float* ao    = (float*)(ws + 2916352);    // 131072 f
  float* a21   = (float*)(ws + 3440640);    // 86016 f  (336 KiB)

  float* out_bond = (float*)d_out;                    // [64,64,128]
  float* out_atom = out_bond + 64 * 64 * 128;         // [64,21]

  k_bond_mlp<<<N_ATOMS * KB / 256, 256, 0, stream>>>(bond_feat,
      F(3), F(4), F(5), F(6), F(7), F(8), bond);
  k_gid<<<N_ATOMS / 256, 256, 0, stream>>>(Mmat, gid);
  k_bond_group<<<GG, 256, 0, stream>>>(bond, gid, bp2);
  k_p2_h1<<<4096 / 256, 256, 0, stream>>>(bp2, F(9), F(10), F(11), F(12), h1buf);
  k_gemm_h1w2<<<2048, 32, 0, stream>>>(h1buf, F(13), F(14), out_bond);

  k_mlp_qkv<<<N_ATOMS / 256, 256, 0, stream>>>(atom_feat,
      F(15), F(16), F(17), F(18), F(19), F(20), qb);
  k_mlp_qkv<<<N_ATOMS / 256, 256, 0, stream>>>(atom_feat,
      F(21), F(22), F(23), F(24), F(25), F(26), kb);
  k_mlp_qkv<<<N_ATOMS / 256, 256, 0, stream>>>(atom_feat,
      F(27), F(28), F(29), F(30), F(31), F(32), vb);
  k_attn<<<2048, 32, 0, stream>>>(qb, kb, vb, ao);
  k_atom_head<<<N_ATOMS / 256, 256, 0, stream>>>(ao,
      F(33), F(34), F(35), F(36), F(37), F(38),
      F(39), F(40), F(41), F(42), F(43), F(44), a21);
  k_atom_pool<<<GG, 256, 0, stream>>>(a21, gid, out_atom);
}